// re_unit_24781961298184
// MI455X (gfx1250) — compile-verified
//
#include <hip/hip_runtime.h>
#include <hip/hip_bf16.h>
#include <math.h>

#define LL 128
#define BB 16
#define HH 256
#define RR 12

typedef __attribute__((ext_vector_type(2))) float v2f;
typedef __attribute__((ext_vector_type(8))) float v8f;

// ---------------------------------------------------------------------------
// GEMM1: re_global[m, n] = tanh( sum_k concat(h_share,h_re)[m,k] * w_r[n,k] + b_r[n] )
// m = l*B+b (2048 rows), K = 512 (concat), N = 256.  One wave = one 16x16 tile.
// V_WMMA_F32_16X16X4_F32: A 16x4 f32 (v2f/lane), B 4x16 f32 (v2f/lane), C 16x16 f32 (v8f).
// ---------------------------------------------------------------------------
__global__ __launch_bounds__(32) void k_gemm_reglobal(
    const float* __restrict__ h_re, const float* __restrict__ h_share,
    const float* __restrict__ w_r, const float* __restrict__ b_r,
    float* __restrict__ re_global)
{
  const int lane = threadIdx.x & 31;
  const int half = lane >> 4;   // 0: K pair (k0,k0+1), 1: (k0+2,k0+3)
  const int lm   = lane & 15;
  const int m0   = blockIdx.x * 16;
  const int n0   = blockIdx.y * 16;

  v8f acc = {};
  const float* wrow = w_r + (size_t)(n0 + lm) * (2 * HH);
  for (int k0 = 0; k0 < 2 * HH; k0 += 4) {
    const float* X = (k0 < HH) ? h_share : h_re;      // concat along K (wave-uniform)
    const int kx = (k0 < HH ? k0 : k0 - HH) + 2 * half;
    v2f av = *(const v2f*)(X + (size_t)(m0 + lm) * HH + kx);      // A[m][k], A[m][k+1]
    v2f bv = *(const v2f*)(wrow + k0 + 2 * half);                 // B[k][n]=w_r[n][k]
    acc = __builtin_amdgcn_wmma_f32_16x16x4_f32(false, av, false, bv,
                                                (short)0, acc, false, false);
  }
  const int n = n0 + lm;
  const float bias = b_r[n];
#pragma unroll
  for (int r = 0; r < 8; ++r) {
    const int mrow = m0 + r + 8 * half;               // C layout: VGPR r -> M=r / M=r+8
    re_global[(size_t)mrow * HH + n] = tanhf(acc[r] + bias);
  }
}

// ---------------------------------------------------------------------------
// GEMM2+3 fused: a = h_re @ W1^T , c = h_re @ W2^T   (share A-fragment loads)
// W1[n][k] = w_hid2hid[n*3H + k], W2[n][k] = w_hid2hid[n*3H + H + k]
// ---------------------------------------------------------------------------
__global__ __launch_bounds__(32) void k_gemm_ac(
    const float* __restrict__ h_re, const float* __restrict__ w_h2h,
    float* __restrict__ a_out, float* __restrict__ c_out)
{
  const int lane = threadIdx.x & 31;
  const int half = lane >> 4;
  const int lm   = lane & 15;
  const int m0   = blockIdx.x * 16;
  const int n0   = blockIdx.y * 16;

  v8f accA = {};
  v8f accC = {};
  const float* w1 = w_h2h + (size_t)(n0 + lm) * (3 * HH);
  const float* w2 = w1 + HH;
  const float* xr = h_re + (size_t)(m0 + lm) * HH;
  for (int k0 = 0; k0 < HH; k0 += 4) {
    const int kk = k0 + 2 * half;
    v2f av = *(const v2f*)(xr + kk);
    v2f b1 = *(const v2f*)(w1 + kk);
    v2f b2 = *(const v2f*)(w2 + kk);
    accA = __builtin_amdgcn_wmma_f32_16x16x4_f32(false, av, false, b1,
                                                 (short)0, accA, false, false);
    accC = __builtin_amdgcn_wmma_f32_16x16x4_f32(false, av, false, b2,
                                                 (short)0, accC, false, false);
  }
  const int n = n0 + lm;
#pragma unroll
  for (int r = 0; r < 8; ++r) {
    const int mrow = m0 + r + 8 * half;
    a_out[(size_t)mrow * HH + n] = accA[r];
    c_out[(size_t)mrow * HH + n] = accC[r];
  }
}

// ---------------------------------------------------------------------------
// h_max[b,h] = max_l re_global[l,b,h]
// ---------------------------------------------------------------------------
__global__ void k_hmax(const float* __restrict__ re_global, float* __restrict__ h_max)
{
  const int idx = blockIdx.x * blockDim.x + threadIdx.x;   // 0 .. B*H-1
  float m = -INFINITY;
  for (int l = 0; l < LL; ++l)
    m = fmaxf(m, re_global[(size_t)l * BB * HH + idx]);
  h_max[idx] = m;
}

// ---------------------------------------------------------------------------
// g[b,n] = h_max[b,:] @ W3[n,:]^T + b_hid2hid[n]   (M=16 -> a single M tile)
// W3[n][k] = w_hid2hid[n*3H + 2H + k]
// ---------------------------------------------------------------------------
__global__ __launch_bounds__(32) void k_gemm_g(
    const float* __restrict__ h_max, const float* __restrict__ w_h2h,
    const float* __restrict__ b_h2h, float* __restrict__ g_out)
{
  const int lane = threadIdx.x & 31;
  const int half = lane >> 4;
  const int lm   = lane & 15;
  const int n0   = blockIdx.x * 16;

  v8f acc = {};
  const float* w3 = w_h2h + (size_t)(n0 + lm) * (3 * HH) + 2 * HH;
  const float* xr = h_max + (size_t)lm * HH;
  for (int k0 = 0; k0 < HH; k0 += 4) {
    const int kk = k0 + 2 * half;
    v2f av = *(const v2f*)(xr + kk);
    v2f bv = *(const v2f*)(w3 + kk);
    acc = __builtin_amdgcn_wmma_f32_16x16x4_f32(false, av, false, bv,
                                                (short)0, acc, false, false);
  }
  const int n = n0 + lm;
  const float bias = b_h2h[n];
#pragma unroll
  for (int r = 0; r < 8; ++r) {
    const int b = r + 8 * half;
    g_out[(size_t)b * HH + n] = acc[r] + bias;
  }
}

// ---------------------------------------------------------------------------
// Fused pair stage: for each task (i,j,b) one wave32 rebuilds
//   pre[h] = a[i,b,h] + c[j,b,h] + g[b,h]        (g already has b_hid2hid)
// does LayerNorm + ELU in-register, projects to R=12 with LDS-cached
// w_hid2rel, applies sigmoid and pair mask.  Never materializes the
// (L,L,B,H) = 268 MB tensor.
// ---------------------------------------------------------------------------
__global__ __launch_bounds__(256) void k_pair_fused(
    const float* __restrict__ a_mat, const float* __restrict__ c_mat,
    const float* __restrict__ g_mat, const int* __restrict__ mask,
    const float* __restrict__ ln_gamma, const float* __restrict__ ln_beta,
    const float* __restrict__ w_rel, const float* __restrict__ b_rel,
    float* __restrict__ out_re)
{
  __shared__ float s_gamma[HH];
  __shared__ float s_beta[HH];
  __shared__ float s_wrel[RR * HH];
  __shared__ float s_brel[RR];

  const int tid = threadIdx.x;
  if (tid < HH) { s_gamma[tid] = ln_gamma[tid]; s_beta[tid] = ln_beta[tid]; }
  for (int t = tid; t < RR * HH; t += blockDim.x) s_wrel[t] = w_rel[t];
  if (tid < RR) s_brel[tid] = b_rel[tid];
  __syncthreads();

  const int lane = tid & 31;
  const int wave = tid >> 5;
  const size_t task = (size_t)blockIdx.x * 8 + wave;   // ((i*L + j)*B + b)
  const int b  = (int)(task % BB);
  const int ij = (int)(task / BB);
  const int j  = ij % LL;
  const int i  = ij / LL;

  const float* ar = a_mat + (size_t)(i * BB + b) * HH;
  const float* cr = c_mat + (size_t)(j * BB + b) * HH;
  const float* gr = g_mat + (size_t)b * HH;

  float pre[8];
  float s1 = 0.f, s2 = 0.f;
#pragma unroll
  for (int u = 0; u < 8; ++u) {
    const int h = lane + 32 * u;                 // coalesced, bank-conflict free
    const float p = ar[h] + cr[h] + gr[h];
    pre[u] = p;
    s1 += p;
    s2 += p * p;
  }
#pragma unroll
  for (int off = 16; off > 0; off >>= 1) {       // wave32 tree reduction
    s1 += __shfl_xor(s1, off, 32);
    s2 += __shfl_xor(s2, off, 32);
  }
  const float inv  = 1.0f / (float)HH;
  const float mean = s1 * inv;
  const float var  = s2 * inv - mean * mean;
  const float rstd = rsqrtf(var + 1e-5f);

  float acc[RR];
#pragma unroll
  for (int r = 0; r < RR; ++r) acc[r] = 0.f;
#pragma unroll
  for (int u = 0; u < 8; ++u) {
    const int h = lane + 32 * u;
    float x = (pre[u] - mean) * rstd * s_gamma[h] + s_beta[h];
    x = (x > 0.f) ? x : expm1f(x);               // ELU(alpha=1)
#pragma unroll
    for (int r = 0; r < RR; ++r) acc[r] += x * s_wrel[r * HH + h];
  }
#pragma unroll
  for (int r = 0; r < RR; ++r) {
#pragma unroll
    for (int off = 16; off > 0; off >>= 1) acc[r] += __shfl_xor(acc[r], off, 32);
  }

  const float pm = (mask[i * BB + b] != 0 && mask[j * BB + b] != 0) ? 1.f : 0.f;
  if (lane == 0) {
    float* o = out_re + task * RR;
#pragma unroll
    for (int r = 0; r < RR; ++r) {
      const float v = acc[r] + s_brel[r];
      o[r] = pm / (1.f + expf(-v));              // pm * sigmoid(v)
    }
  }
}

// ---------------------------------------------------------------------------
extern "C" void kernel_launch(void* const* d_in, const int* in_sizes, int n_in,
                              void* d_out, int out_size, void* d_ws, size_t ws_size,
                              hipStream_t stream)
{
  (void)in_sizes; (void)n_in; (void)out_size; (void)ws_size;

  const float* h_re    = (const float*)d_in[0];   // (L,B,H)
  const float* h_share = (const float*)d_in[1];   // (L,B,H)
  const int*   mask    = (const int*)  d_in[2];   // (L,B)
  const float* w_r     = (const float*)d_in[3];   // (H, 2H)
  const float* b_r     = (const float*)d_in[4];   // (H,)
  const float* w_h2h   = (const float*)d_in[5];   // (H, 3H)
  const float* b_h2h   = (const float*)d_in[6];   // (H,)
  const float* gamma   = (const float*)d_in[7];   // (H,)
  const float* beta    = (const float*)d_in[8];   // (H,)
  const float* w_rel   = (const float*)d_in[9];   // (R, H)
  const float* b_rel   = (const float*)d_in[10];  // (R,)

  float* out_re       = (float*)d_out;                         // (L,L,B,R)
  float* out_reglobal = out_re + (size_t)LL * LL * BB * RR;    // (L,B,H)

  float* ws    = (float*)d_ws;
  float* a_mat = ws;                                  // (L,B,H)  2 MB
  float* c_mat = a_mat + (size_t)LL * BB * HH;        // (L,B,H)  2 MB
  float* h_max = c_mat + (size_t)LL * BB * HH;        // (B,H)   16 KB
  float* g_mat = h_max + (size_t)BB * HH;             // (B,H)   16 KB

  // 1) re_global = tanh(concat @ w_r^T + b_r)   [WMMA f32]
  k_gemm_reglobal<<<dim3(LL * BB / 16, HH / 16), 32, 0, stream>>>(
      h_re, h_share, w_r, b_r, out_reglobal);
  // 2) a, c GEMMs (shared A fragments)          [WMMA f32]
  k_gemm_ac<<<dim3(LL * BB / 16, HH / 16), 32, 0, stream>>>(
      h_re, w_h2h, a_mat, c_mat);
  // 3) h_max over L
  k_hmax<<<dim3(BB * HH / 256), 256, 0, stream>>>(out_reglobal, h_max);
  // 4) g = h_max @ W3^T + b_hid2hid             [WMMA f32]
  k_gemm_g<<<dim3(HH / 16), 32, 0, stream>>>(h_max, w_h2h, b_h2h, g_mat);
  // 5) fused LN + ELU + projection + sigmoid + pair mask (one wave per (i,j,b))
  k_pair_fused<<<dim3(LL * LL * BB / 8), 256, 0, stream>>>(
      a_mat, c_mat, g_mat, mask, gamma, beta, w_rel, b_rel, out_re);
}